// loss_with_array_20564303413528
// MI455X (gfx1250) — compile-verified
//
#include <hip/hip_runtime.h>
#include <hip/hip_bf16.h>
#include <math.h>

typedef __attribute__((ext_vector_type(2))) float v2f;
typedef __attribute__((ext_vector_type(8))) float v8f;

#define L_SAMP 128
#define N_TOT  65536
#define D_DIM  8
#define DZ_DIM 16
#define NPB    16                 // n's handled per block (2 per wave * 8 waves)
#define NTHREADS 256
#define NWAVES (NTHREADS / 32)
#define NBLOCKS (N_TOT / NPB)     // 4096

// ---------------------------------------------------------------------------
// Main kernel: one streaming pass over y_pred (256 MB).
// Each wave owns two adjacent n's. S2 = sum_l y y^T accumulated with
// v_wmma_f32_16x16x4_f32 (two 8x8 Gram blocks packed in one 16x16 tile),
// S1 = sum_l y accumulated in VALU. Then 16 threads/block do the 8x8
// Cholesky (logdet + quadratic form) and 256 threads do the KL elements.
// ---------------------------------------------------------------------------
__global__ __launch_bounds__(NTHREADS)
void cov_loss_main(const float* __restrict__ y_true,
                   const float* __restrict__ y_pred,
                   const float* __restrict__ mu,
                   const float* __restrict__ Sigma,
                   float* __restrict__ partial)
{
  __shared__ float sh_S2[NPB][D_DIM][D_DIM + 1];  // +1 pad vs bank conflicts
  __shared__ float sh_S1[NPB][D_DIM];
  __shared__ float sh_red[NWAVES][2];

  const int tid   = threadIdx.x;
  const int wave  = tid >> 5;       // 0..7
  const int lane  = tid & 31;
  const int khalf = lane >> 4;      // 0: K=0,1 ; 1: K=2,3 (ISA A/B layout)
  const int li    = lane & 15;      // row/col index M = N = li
  const int d     = li & 7;         // dim within a covariance
  const int wn    = li >> 3;        // which n of the wave's pair

  const int n_blk = blockIdx.x * NPB;
  const int n     = n_blk + wave * 2 + wn;

  const size_t rowstride = (size_t)N_TOT * D_DIM;      // floats between l's
  const float* base = y_pred + (size_t)n * D_DIM + d;

  v8f   c  = {};      // 16x16 f32 accumulator (8 VGPRs)
  float s1 = 0.0f;

  for (int k = 0; k < L_SAMP / 4; ++k) {
    const int l0 = 4 * k + 2 * khalf;
    const float* p = base + (size_t)l0 * rowstride;
    if (k + 1 < L_SAMP / 4) __builtin_prefetch(p + 4 * rowstride, 0, 1);
    v2f ab;                       // identical A and B image for X^T X
    ab.x = p[0];                  // K = l0
    ab.y = p[rowstride];          // K = l0 + 1
    s1 += ab.x + ab.y;
    c = __builtin_amdgcn_wmma_f32_16x16x4_f32(
            /*neg_a=*/false, ab, /*neg_b=*/false, ab,
            /*c_mod=*/(short)0, c, /*reuse_a=*/false, /*reuse_b=*/false);
  }

  // S1: lanes (i) and (i+16) cover complementary halves of l
  s1 += __shfl_xor(s1, 16, 32);
  if (khalf == 0) sh_S1[wave * 2 + wn][d] = s1;

  // C layout: lanes 0-7 hold n_even's 8x8 (col=lane, row=VGPR);
  //           lanes 24-31 hold n_odd's 8x8 (col=lane-24, row=VGPR).
  {
    int ln = -1, col = 0;
    if (lane < 8)        { ln = wave * 2;     col = lane;      }
    else if (lane >= 24) { ln = wave * 2 + 1; col = lane - 24; }
    if (ln >= 0) {
#pragma unroll
      for (int r = 0; r < D_DIM; ++r) sh_S2[ln][r][col] = c[r];
    }
  }
  __syncthreads();

  // ---- per-n reconstruction term: 0.5*(log|det C| + m^T C^-1 m) ----
  float rec = 0.0f;
  if (tid < NPB) {
    const float invL   = 1.0f / (float)L_SAMP;
    const float invLm1 = 1.0f / (float)(L_SAMP - 1);
    const int   ng     = n_blk + tid;

    float S1v[D_DIM], mv[D_DIM];
#pragma unroll
    for (int i = 0; i < D_DIM; ++i) {
      S1v[i] = sh_S1[tid][i];
      mv[i]  = S1v[i] * invL - y_true[(size_t)ng * D_DIM + i];
    }

    float Lm[D_DIM][D_DIM];
    float logdet = 0.0f;
#pragma unroll
    for (int j = 0; j < D_DIM; ++j) {
      float diag = (sh_S2[tid][j][j] - S1v[j] * S1v[j] * invL) * invLm1;
#pragma unroll
      for (int p = 0; p < D_DIM; ++p)
        if (p < j) diag -= Lm[j][p] * Lm[j][p];
      logdet += logf(diag);
      const float sj  = sqrtf(diag);
      const float inv = 1.0f / sj;
      Lm[j][j] = sj;
#pragma unroll
      for (int i = 0; i < D_DIM; ++i) {
        if (i > j) {
          float v = (sh_S2[tid][i][j] - S1v[i] * S1v[j] * invL) * invLm1;
#pragma unroll
          for (int p = 0; p < D_DIM; ++p)
            if (p < j) v -= Lm[i][p] * Lm[j][p];
          Lm[i][j] = v * inv;
        }
      }
    }
    // forward solve L z = m ; quad = z^T z = m^T C^-1 m
    float z[D_DIM];
    float quad = 0.0f;
#pragma unroll
    for (int i = 0; i < D_DIM; ++i) {
      float v = mv[i];
#pragma unroll
      for (int p = 0; p < D_DIM; ++p)
        if (p < i) v -= Lm[i][p] * z[p];
      v /= Lm[i][i];
      z[i] = v;
      quad += v * v;
    }
    rec = 0.5f * (logdet + quad);
  }

  // ---- KL: one (n, dz) element per thread (16 n * 16 dz = 256) ----
  float kl;
  {
    const int   nk  = n_blk + (tid >> 4);
    const int   dim = tid & 15;
    const float S   = Sigma[(size_t)nk * DZ_DIM + dim];
    const float m   = mu[(size_t)nk * DZ_DIM + dim];
    kl = -0.5f * (1.0f + logf(S) - m * m - S);
  }

  // ---- deterministic block reduction ----
  float r0 = rec, r1 = kl;
#pragma unroll
  for (int off = 16; off > 0; off >>= 1) {
    r0 += __shfl_down(r0, off, 32);
    r1 += __shfl_down(r1, off, 32);
  }
  if (lane == 0) { sh_red[wave][0] = r0; sh_red[wave][1] = r1; }
  __syncthreads();
  if (tid == 0) {
    float pr = 0.0f, pk = 0.0f;
#pragma unroll
    for (int w = 0; w < NWAVES; ++w) { pr += sh_red[w][0]; pk += sh_red[w][1]; }
    partial[2 * (size_t)blockIdx.x + 0] = pr;
    partial[2 * (size_t)blockIdx.x + 1] = pk;
  }
}

// ---------------------------------------------------------------------------
// Finalize: reduce per-block partials, normalize by N, emit 3 scalars.
// ---------------------------------------------------------------------------
__global__ __launch_bounds__(256)
void cov_loss_finalize(const float* __restrict__ partial,
                       float* __restrict__ out, int nblocks)
{
  __shared__ float sh[8][2];
  const int tid = threadIdx.x, lane = tid & 31, wave = tid >> 5;
  float r = 0.0f, k = 0.0f;
  for (int i = tid; i < nblocks; i += 256) {
    r += partial[2 * (size_t)i + 0];
    k += partial[2 * (size_t)i + 1];
  }
#pragma unroll
  for (int off = 16; off > 0; off >>= 1) {
    r += __shfl_down(r, off, 32);
    k += __shfl_down(k, off, 32);
  }
  if (lane == 0) { sh[wave][0] = r; sh[wave][1] = k; }
  __syncthreads();
  if (tid == 0) {
    float pr = 0.0f, pk = 0.0f;
#pragma unroll
    for (int w = 0; w < 8; ++w) { pr += sh[w][0]; pk += sh[w][1]; }
    const float invN = 1.0f / (float)N_TOT;
    const float Rec = pr * invN;
    const float KL  = pk * invN;
    out[0] = KL + Rec;
    out[1] = KL;
    out[2] = Rec;
  }
}

extern "C" void kernel_launch(void* const* d_in, const int* in_sizes, int n_in,
                              void* d_out, int out_size, void* d_ws, size_t ws_size,
                              hipStream_t stream) {
  (void)in_sizes; (void)n_in; (void)out_size; (void)ws_size;
  const float* y_true = (const float*)d_in[0];   // (N, D)
  const float* y_pred = (const float*)d_in[1];   // (L, N, D)
  const float* mu     = (const float*)d_in[2];   // (N, DZ)
  const float* Sigma  = (const float*)d_in[3];   // (N, DZ)
  float* out     = (float*)d_out;                // 3 floats
  float* partial = (float*)d_ws;                 // NBLOCKS * 2 floats

  cov_loss_main<<<NBLOCKS, NTHREADS, 0, stream>>>(y_true, y_pred, mu, Sigma, partial);
  cov_loss_finalize<<<1, 256, 0, stream>>>(partial, out, NBLOCKS);
}